// ROIFeatureFusion_7636451852912
// MI455X (gfx1250) — compile-verified
//
#include <hip/hip_runtime.h>
#include <hip/hip_bf16.h>

// CDNA5 (gfx1250) wave32 WMMA types
typedef __attribute__((ext_vector_type(2))) float v2f;
typedef __attribute__((ext_vector_type(8))) float v8f;

#define SZ 512
#define CC 81      // channels
#define HM 28      // mask size
#define NB 128     // boxes

// Workspace layout (floats):
//   att  : [512*512]                          offset 0
//   w3f  : [9][21][6][32][2] = 72576          offset 262144
//   w1f  : [21][6][32][2]    = 8064           offset 334720
//   b3p  : [96]                               offset 342784
//   b1p  : [96]                               offset 342880
#define WS_ATT 0
#define WS_W3F 262144
#define WS_W1F 334720
#define WS_B3P 342784
#define WS_B1P 342880
#define N_W3F  72576
#define N_W1F  8064

// ---------------------------------------------------------------------------
// Kernel 0: build zero-padded, WMMA-fragment-ordered weight/bias tables.
// Fragment layout matches V_WMMA_F32_16X16X4_F32 A-operand:
//   lane<16 : v2f = { W[k+0][co], W[k+1][co] },  lane>=16 : { W[k+2][co], W[k+3][co] }
//   with co = cotile*16 + (lane&15), k = kb*4.
// ---------------------------------------------------------------------------
__global__ __launch_bounds__(256) void prep_kernel(const float* __restrict__ w3,
                                                   const float* __restrict__ b3,
                                                   const float* __restrict__ w1,
                                                   const float* __restrict__ b1,
                                                   float* __restrict__ ws) {
  int idx = blockIdx.x * 256 + threadIdx.x;
  if (idx < N_W3F) {
    int within = idx & 63;
    int lane = within >> 1, j = within & 1;
    int f = idx >> 6;
    int ct = f % 6; f /= 6;
    int kb = f % 21;
    int t  = f / 21;
    int co = ct * 16 + (lane & 15);
    int k  = kb * 4 + ((lane >= 16) ? 2 : 0) + j;
    ws[WS_W3F + idx] = (co < CC && k < CC) ? w3[co * (CC * 9) + k * 9 + t] : 0.0f;
  } else if (idx < N_W3F + N_W1F) {
    int i2 = idx - N_W3F;
    int within = i2 & 63;
    int lane = within >> 1, j = within & 1;
    int f = i2 >> 6;
    int ct = f % 6;
    int kb = f / 6;
    int co = ct * 16 + (lane & 15);
    int k  = kb * 4 + ((lane >= 16) ? 2 : 0) + j;
    ws[WS_W1F + i2] = (co < CC && k < CC) ? w1[co * CC + k] : 0.0f;
  } else if (idx < N_W3F + N_W1F + 96) {
    int c = idx - N_W3F - N_W1F;
    ws[WS_B3P + c] = (c < CC) ? b3[c] : 0.0f;
  } else if (idx < N_W3F + N_W1F + 192) {
    int c = idx - N_W3F - N_W1F - 96;
    ws[WS_B1P + c] = (c < CC) ? b1[c] : 0.0f;
  }
}

// ---------------------------------------------------------------------------
// Kernel 1: multiplicative box-attention map, one thread per pixel.
// ---------------------------------------------------------------------------
__global__ __launch_bounds__(256) void att_kernel(const float* __restrict__ masks,
                                                  const int*   __restrict__ boxes,
                                                  float*       __restrict__ att) {
  int idx = blockIdx.x * blockDim.x + threadIdx.x;   // 0 .. 512*512-1
  int r = idx >> 9;
  int c = idx & (SZ - 1);
  float prod = 1.0f;
  for (int n = 0; n < NB; ++n) {
    int x0 = boxes[n * 4 + 0];
    int y0 = boxes[n * 4 + 1];
    int x1 = boxes[n * 4 + 2];
    int y1 = boxes[n * 4 + 3];
    int wi = max(x1 - x0 + 1, 1);
    int hi = max(y1 - y0 + 1, 1);
    int dy = r - y0;
    int dx = c - x0;
    if (dy >= 0 && dy < wi && dx >= 0 && dx < hi) {
      float sy = fminf(fmaxf(((float)dy + 0.5f) * ((float)HM / (float)wi) - 0.5f, 0.0f), (float)(HM - 1));
      float sx = fminf(fmaxf(((float)dx + 0.5f) * ((float)HM / (float)hi) - 0.5f, 0.0f), (float)(HM - 1));
      int   yi = (int)floorf(sy);
      float fy = sy - (float)yi;
      int   yj = min(yi + 1, HM - 1);
      int   xi = (int)floorf(sx);
      float fx = sx - (float)xi;
      int   xj = min(xi + 1, HM - 1);
      const float* m = masks + n * (HM * HM);
      float v00 = m[yi * HM + xi];
      float v01 = m[yi * HM + xj];
      float v10 = m[yj * HM + xi];
      float v11 = m[yj * HM + xj];
      float a   = v00 * (1.0f - fy) + v10 * fy;
      float b   = v01 * (1.0f - fy) + v11 * fy;
      float val = a * (1.0f - fx) + b * fx;
      float sig = 1.0f / (1.0f + __expf(-val));
      prod *= (1.0f + sig);
    }
  }
  att[idx] = prod;
}

// ---------------------------------------------------------------------------
// Kernel 2: fused  relu(fm*att) -> conv3x3(+b3) -> relu -> conv1x1(+b1)
// Block = 256 threads = 8 waves; tile = 16 wide x 4 high pixels, all channels.
//   wave w: px row pw = w&3, channel-set cset = w>>2 (co tiles cset*3..cset*3+2)
// LDS tiles are stored as K-PAIRS (parity innermost) so every WMMA B operand
// is ONE contiguous, 8B-aligned ds_load_b64 landing in an even VGPR pair:
//   sX2[(ci>>1)*112 + plane(6x18)][ci&1], plane padded 108->112 so the two
//   half-waves (pair vs pair+1) hit disjoint LDS bank arcs (2*112 % 64 == 32).
// 3 accumulators/wave keeps VGPR usage < 256 (no S_SET_VGPR_MSB churn) and
// lets the scheduler keep ~40 loads in flight ahead of the WMMA pipe.
// ---------------------------------------------------------------------------
__global__ __launch_bounds__(256) void conv_kernel(const float* __restrict__ fm,
                                                   const float* __restrict__ ws,
                                                   float*       __restrict__ out) {
  __shared__ __align__(16) float sAtt[6 * 18];     // attention halo patch
  __shared__ __align__(16) float sX2[42 * 112 * 2];// relu(fm*att), ci-paired
  __shared__ __align__(16) float sY2[48 * 64 * 2]; // conv3 out relu'd, co-paired

  const float* att = ws + WS_ATT;
  const float* w3f = ws + WS_W3F;
  const float* w1f = ws + WS_W1F;
  const float* b3p = ws + WS_B3P;
  const float* b1p = ws + WS_B1P;

  const int bx  = blockIdx.x;             // 0..31  (col block,  16 px)
  const int by  = blockIdx.y;             // 0..127 (row block,   4 px)
  const int tid = threadIdx.x;
  const int lane = tid & 31;
  const int wv   = tid >> 5;

  // ---- stage attention patch (6x18 with halo) ----
  for (int i = tid; i < 6 * 18; i += 256) {
    int pr = i / 18, pc = i % 18;
    int r = by * 4 + pr - 1;
    int c = bx * 16 + pc - 1;
    sAtt[i] = (r >= 0 && r < SZ && c >= 0 && c < SZ) ? att[r * SZ + c] : 0.0f;
  }
  __syncthreads();

  // ---- stage X = relu(fm * att) halo patch; ci rows 81..83 stay zero ----
  for (int i = tid; i < 84 * 108; i += 256) {
    int ci  = i / 108;
    int rem = i % 108;                    // pr*18 + pc within the 6x18 plane
    int pr = rem / 18, pc = rem % 18;
    int r = by * 4 + pr - 1;
    int c = bx * 16 + pc - 1;
    float v = 0.0f;
    if (ci < CC && r >= 0 && r < SZ && c >= 0 && c < SZ) {
      v = fmaxf(fm[ci * (SZ * SZ) + r * SZ + c] * sAtt[rem], 0.0f);
    }
    sX2[((ci >> 1) * 112 + rem) * 2 + (ci & 1)] = v;
  }
  __syncthreads();

  const int pw    = wv & 3;               // pixel row within the 16x4 tile
  const int cset  = wv >> 2;              // 0/1 -> co tiles [cset*3 .. cset*3+2]
  const int hi16  = (lane >= 16) ? 1 : 0;
  const int l16   = lane & 15;

  // per-wave bases (computed once)
  const float* w3w = w3f + (cset * 3) * 64 + lane * 2;   // + (t*21+kb)*384 + tj*64
  const float* w1w = w1f + (cset * 3) * 64 + lane * 2;   // + kb*384 + tj*64

  // ---- GEMM 1: conv3x3.  M=co(16), N=px(16), K=ci (81 pad 84) per tap ----
  v8f acc[3] = {};
#pragma unroll
  for (int t = 0; t < 9; ++t) {
    const int ky = t / 3, kx = t % 3;                    // constants after unroll
    // pair index for K-step kb at this half-wave: kb*2 + hi16  (K = kb*4+2*hi16)
    const float* xrow = &sX2[hi16 * 224 + ((pw + ky) * 18 + l16 + kx) * 2];
#pragma unroll
    for (int kb = 0; kb < 21; ++kb) {
      v2f b = *(const v2f*)(xrow + kb * 448);            // one ds_load_b64
      const float* wp = w3w + (t * 21 + kb) * 384;
#pragma unroll
      for (int tj = 0; tj < 3; ++tj) {
        v2f a = *(const v2f*)(wp + tj * 64);             // global_load_b64, L2-hit
        acc[tj] = __builtin_amdgcn_wmma_f32_16x16x4_f32(
            false, a, false, b, (short)0, acc[tj], false, false);
      }
    }
  }

  // ---- epilogue 1: +b3, relu -> sY2 co-paired (pad rows forced to 0) ----
#pragma unroll
  for (int tj = 0; tj < 3; ++tj) {
    const int cobase = (cset * 3 + tj) * 16;
#pragma unroll
    for (int v = 0; v < 8; ++v) {
      const int co = cobase + v + (hi16 ? 8 : 0);
      const int px = pw * 16 + l16;
      sY2[((co >> 1) * 64 + px) * 2 + (co & 1)] = fmaxf(acc[tj][v] + b3p[co], 0.0f);
    }
  }
  __syncthreads();

  // ---- GEMM 2: conv1x1.  M=co2(16), N=px(16), K=co (81 pad 84) ----
  v8f acc2[3] = {};
  const float* yb = &sY2[hi16 * 128 + (pw * 16 + l16) * 2];
#pragma unroll
  for (int kb = 0; kb < 21; ++kb) {
    v2f b = *(const v2f*)(yb + kb * 256);                // one ds_load_b64
    const float* wp = w1w + kb * 384;
#pragma unroll
    for (int tj = 0; tj < 3; ++tj) {
      v2f a = *(const v2f*)(wp + tj * 64);
      acc2[tj] = __builtin_amdgcn_wmma_f32_16x16x4_f32(
          false, a, false, b, (short)0, acc2[tj], false, false);
    }
  }

  // ---- epilogue 2: +b1, store (no final relu, matches reference) ----
  const int r = by * 4 + pw;
  const int c = bx * 16 + l16;
#pragma unroll
  for (int tj = 0; tj < 3; ++tj) {
    const int cobase = (cset * 3 + tj) * 16;
#pragma unroll
    for (int v = 0; v < 8; ++v) {
      const int co2 = cobase + v + (hi16 ? 8 : 0);
      if (co2 < CC) {
        out[co2 * (SZ * SZ) + r * SZ + c] = acc2[tj][v] + b1p[co2];
      }
    }
  }
}

extern "C" void kernel_launch(void* const* d_in, const int* in_sizes, int n_in,
                              void* d_out, int out_size, void* d_ws, size_t ws_size,
                              hipStream_t stream) {
  const float* masks = (const float*)d_in[0];   // [128,28,28] f32
  const int*   boxes = (const int*)  d_in[1];   // [128,4] i32
  const float* fm    = (const float*)d_in[2];   // [1,81,512,512] f32
  const float* w3    = (const float*)d_in[3];   // [81,81,3,3] f32
  const float* b3    = (const float*)d_in[4];   // [81] f32
  const float* w1    = (const float*)d_in[5];   // [81,81,1,1] f32
  const float* b1    = (const float*)d_in[6];   // [81] f32
  float* out = (float*)d_out;                   // [1,81,512,512] f32
  float* ws  = (float*)d_ws;                    // ~1.4 MB used

  prep_kernel<<<(N_W3F + N_W1F + 192 + 255) / 256, 256, 0, stream>>>(w3, b3, w1, b1, ws);
  att_kernel<<<(SZ * SZ) / 256, 256, 0, stream>>>(masks, boxes, ws + WS_ATT);

  dim3 grid(SZ / 16, SZ / 4);                   // 32 x 128 blocks
  conv_kernel<<<grid, 256, 0, stream>>>(fm, ws, out);
}